// SpGATLayer_5171140625173
// MI455X (gfx1250) — compile-verified
//
#include <hip/hip_runtime.h>

typedef __attribute__((ext_vector_type(2))) float v2f;
typedef __attribute__((ext_vector_type(8))) float v8f;

#define N_NODES 50000
#define N_EDGES 800000
#define N_HEAD  4
#define F_IN    256
#define F_OUT   64

__device__ __forceinline__ void atomic_add_f32(float* p, float v) {
    __hip_atomic_fetch_add(p, v, __ATOMIC_RELAXED, __HIP_MEMORY_SCOPE_AGENT);
}

// ---------------------------------------------------------------------------
// Zero-fill (grid-stride). Harness poisons d_out/d_ws with 0xAA.
// ---------------------------------------------------------------------------
__global__ void zero_kernel(float* __restrict__ a, int n) {
    int i = blockIdx.x * blockDim.x + threadIdx.x;
    int stride = gridDim.x * blockDim.x;
    for (; i < n; i += stride) a[i] = 0.0f;
}

// ---------------------------------------------------------------------------
// Kernel 1: h_prime[n][head][o] = sum_k h[n][k] * w[head][k][o]
// One block = 16 node-rows x one head. 4 waves, each wave owns a 16-wide
// column slice of F_OUT=64. K-loop of 64 x V_WMMA_F32_16X16X4_F32.
//
// A (16x4 f32) per-lane: lane = half*16 + r ; v0 = h[n0+r][k0+2*half],
//                                             v1 = h[n0+r][k0+2*half+1]
// B (4x16 f32) per-lane: v0 = w[k0+2*half][o0+r], v1 = w[k0+2*half+1][o0+r]
// C/D (16x16 f32): c[g] is row M = g + 8*half, col N = r.
// ---------------------------------------------------------------------------
__global__ __launch_bounds__(128) void gemm_hprime_kernel(
    const float* __restrict__ h, const float* __restrict__ w,
    float* __restrict__ h_prime, int n_nodes)
{
    const int n0   = blockIdx.x * 16;
    const int head = blockIdx.y;
    const int wave = threadIdx.x >> 5;
    const int lane = threadIdx.x & 31;
    const int half = lane >> 4;
    const int r    = lane & 15;
    const int o0   = wave * 16;

    int rowA = n0 + r;
    if (rowA >= n_nodes) rowA = n_nodes - 1;   // address clamp only; EXEC stays all-1s

    const float* Aptr = h + (size_t)rowA * F_IN + 2 * half;
    const float* Bptr = w + (size_t)head * (F_IN * F_OUT) + (2 * half) * F_OUT + (o0 + r);

    v8f c = {};
#pragma unroll 8
    for (int k0 = 0; k0 < F_IN; k0 += 4) {
        v2f a = *(const v2f*)(Aptr + k0);      // 8B-aligned float2
        const float* bp = Bptr + (size_t)k0 * F_OUT;
        v2f b;
        b[0] = bp[0];
        b[1] = bp[F_OUT];
        c = __builtin_amdgcn_wmma_f32_16x16x4_f32(
                /*neg_a=*/false, a, /*neg_b=*/false, b,
                /*c_mod=*/(short)0, c, /*reuse_a=*/false, /*reuse_b=*/false);
    }

    // Epilogue: N_NODES % 16 == 0 here, so the whole tile is in range and the
    // fast path issues an unpredicated clause of 8 stores (no exec juggling).
    float* outp = h_prime + ((size_t)(n0 + 8 * half) * N_HEAD + head) * F_OUT + o0 + r;
    if (n0 + 15 < n_nodes) {
#pragma unroll
        for (int g = 0; g < 8; ++g)
            outp[(size_t)g * N_HEAD * F_OUT] = c[g];
    } else {
#pragma unroll
        for (int g = 0; g < 8; ++g) {
            int row = n0 + g + 8 * half;
            if (row < n_nodes)
                h_prime[((size_t)row * N_HEAD + head) * F_OUT + o0 + r] = c[g];
        }
    }
}

// ---------------------------------------------------------------------------
// Kernel 2: per-node attention scalars.
// One 256-thread block per node; thread t reads h_prime[n][t] (coalesced),
// t/64 = head, t%64 = o. Shuffle-reduce within wave32, combine pairs in LDS.
// ---------------------------------------------------------------------------
__global__ __launch_bounds__(256) void attn_kernel(
    const float* __restrict__ h_prime,
    const float* __restrict__ p_src,   // [H][F_OUT] flat = 256
    const float* __restrict__ p_trg,
    float* __restrict__ attn_src,      // [N][H]
    float* __restrict__ attn_trg)
{
    __shared__ float sS[8];
    __shared__ float sT[8];
    const int n = blockIdx.x;
    const int t = threadIdx.x;

    float v  = h_prime[(size_t)n * (N_HEAD * F_OUT) + t];
    float s  = v * p_src[t];
    float tr = v * p_trg[t];
#pragma unroll
    for (int off = 16; off > 0; off >>= 1) {
        s  += __shfl_xor(s,  off, 32);
        tr += __shfl_xor(tr, off, 32);
    }
    if ((t & 31) == 0) { sS[t >> 5] = s; sT[t >> 5] = tr; }
    __syncthreads();
    if (t < N_HEAD) {
        attn_src[n * N_HEAD + t] = sS[2 * t] + sS[2 * t + 1];
        attn_trg[n * N_HEAD + t] = sT[2 * t] + sT[2 * t + 1];
    }
}

// ---------------------------------------------------------------------------
// Kernel 3: per-edge logits -> exp, and denom accumulation.
// Global-max subtraction in the reference cancels in the softmax ratio, so
// it is omitted (values are O(1) here).
// ---------------------------------------------------------------------------
__global__ __launch_bounds__(256) void edge_exp_kernel(
    const float* __restrict__ attn_src, const float* __restrict__ attn_trg,
    const int* __restrict__ src, const int* __restrict__ trg,
    float* __restrict__ exp_e, float* __restrict__ denom, int n_edges)
{
    int e = blockIdx.x * blockDim.x + threadIdx.x;
    if (e >= n_edges) return;
    int s = src[e], t = trg[e];
    float4 as = *(const float4*)(attn_src + 4 * s);
    float4 at = *(const float4*)(attn_trg + 4 * t);
    float x0 = as.x + at.x, x1 = as.y + at.y, x2 = as.z + at.z, x3 = as.w + at.w;
    // leaky_relu(x, 0.2) == max(x, 0.2*x)
    x0 = fmaxf(x0, 0.2f * x0); x1 = fmaxf(x1, 0.2f * x1);
    x2 = fmaxf(x2, 0.2f * x2); x3 = fmaxf(x3, 0.2f * x3);
    float4 ex;
    ex.x = __expf(x0); ex.y = __expf(x1); ex.z = __expf(x2); ex.w = __expf(x3);
    *(float4*)(exp_e + (size_t)e * 4) = ex;
    float* d = denom + (size_t)t * 4;
    atomic_add_f32(d + 0, ex.x);
    atomic_add_f32(d + 1, ex.y);
    atomic_add_f32(d + 2, ex.z);
    atomic_add_f32(d + 3, ex.w);
}

// ---------------------------------------------------------------------------
// Kernel 4: weighted aggregation. One wave32 per edge; each lane handles 8
// of the 256 (head,o) elements in coalesced 128B strides; f32 HW atomics.
// ---------------------------------------------------------------------------
__global__ __launch_bounds__(256) void aggregate_kernel(
    const float* __restrict__ h_prime, const float* __restrict__ exp_e,
    const float* __restrict__ denom,
    const int* __restrict__ src, const int* __restrict__ trg,
    float* __restrict__ out, int n_edges)
{
    int e = blockIdx.x * 8 + (threadIdx.x >> 5);
    if (e >= n_edges) return;
    int lane = threadIdx.x & 31;
    int s = src[e], t = trg[e];

    const float* ee = exp_e + (size_t)e * 4;
    const float* dn = denom + (size_t)t * 4;
    float coef[N_HEAD];
#pragma unroll
    for (int hh = 0; hh < N_HEAD; ++hh)
        coef[hh] = ee[hh] / (dn[hh] + 1e-16f);

    const float* hp = h_prime + (size_t)s * (N_HEAD * F_OUT);
    float*       op = out     + (size_t)t * (N_HEAD * F_OUT);
#pragma unroll
    for (int i = 0; i < 8; ++i) {
        int j = lane + 32 * i;        // head = i>>1 since lane < 32
        atomic_add_f32(op + j, coef[i >> 1] * hp[j]);
    }
}

// ---------------------------------------------------------------------------
// Launch
// ---------------------------------------------------------------------------
extern "C" void kernel_launch(void* const* d_in, const int* in_sizes, int n_in,
                              void* d_out, int out_size, void* d_ws, size_t ws_size,
                              hipStream_t stream) {
    const float* h   = (const float*)d_in[0];   // [N, F_IN]
    const float* w   = (const float*)d_in[1];   // [H, F_IN, F_OUT]
    const float* ps  = (const float*)d_in[2];   // [H, F_OUT, 1]
    const float* pt  = (const float*)d_in[3];   // [H, F_OUT, 1]
    const int*   src = (const int*)d_in[4];     // [E]
    const int*   trg = (const int*)d_in[5];     // [E]
    float* out = (float*)d_out;                 // [N, H, F_OUT]

    const int n_nodes = in_sizes[0] / F_IN;     // 50000
    const int n_edges = in_sizes[4];            // 800000

    // Workspace carve-up (floats): h_prime | attn_src | attn_trg | denom | exp_e
    float* ws       = (float*)d_ws;
    float* h_prime  = ws;                                        // N*H*F_OUT
    float* attn_src = h_prime  + (size_t)n_nodes * N_HEAD * F_OUT;
    float* attn_trg = attn_src + (size_t)n_nodes * N_HEAD;
    float* denom    = attn_trg + (size_t)n_nodes * N_HEAD;
    float* exp_e    = denom    + (size_t)n_nodes * N_HEAD;       // E*H

    zero_kernel<<<1024, 256, 0, stream>>>(out, n_nodes * N_HEAD * F_OUT);
    zero_kernel<<<256, 256, 0, stream>>>(denom, n_nodes * N_HEAD);

    dim3 ggrid((n_nodes + 15) / 16, N_HEAD);
    gemm_hprime_kernel<<<ggrid, 128, 0, stream>>>(h, w, h_prime, n_nodes);

    attn_kernel<<<n_nodes, 256, 0, stream>>>(h_prime, ps, pt, attn_src, attn_trg);

    edge_exp_kernel<<<(n_edges + 255) / 256, 256, 0, stream>>>(
        attn_src, attn_trg, src, trg, exp_e, denom, n_edges);

    aggregate_kernel<<<(n_edges + 7) / 8, 256, 0, stream>>>(
        h_prime, exp_e, denom, src, trg, out, n_edges);
}